// DotInteraction_11974368821390
// MI455X (gfx1250) — compile-verified
//
#include <hip/hip_runtime.h>
#include <stdint.h>

// CDNA5 / gfx1250 — wave32, WMMA + Tensor Data Mover.
typedef __attribute__((ext_vector_type(2))) float    v2f;
typedef __attribute__((ext_vector_type(8))) float    v8f;
typedef __attribute__((ext_vector_type(4))) uint32_t u32x4;
typedef __attribute__((ext_vector_type(4))) int      i32x4;
typedef __attribute__((ext_vector_type(8))) int      i32x8;

#define K_ROWS     64          // rows per batch (K in the reference)
#define D_DIM      128         // feature dim
#define LDS_STRIDE 132         // 128 + 4 floats pad -> fragment rows hit distinct LDS banks
#define NPAIR      2016        // 64*63/2 strictly-lower-tri entries
#define NTHREADS   320         // 10 wave32 waves: one per lower-tri 16x16 tile

#ifndef __has_builtin
#define __has_builtin(x) 0
#endif

#if defined(__AMDGCN__) && __has_builtin(__builtin_amdgcn_tensor_load_to_lds) && \
    __has_builtin(__builtin_amdgcn_s_wait_tensorcnt)
#define HAVE_TDM 1
#else
#define HAVE_TDM 0
#endif

__global__ __launch_bounds__(NTHREADS)
void dot_interaction_wmma_f32(const float* __restrict__ A, float* __restrict__ out)
{
    // 64 rows x 132 floats = 33792 B of LDS (WGP has 320 KB)
    __shared__ float lds[K_ROWS * LDS_STRIDE];

    const int b   = blockIdx.x;
    const int tid = threadIdx.x;
    const float* __restrict__ Ab = A + (size_t)b * (K_ROWS * D_DIM);

#if HAVE_TDM
    // ---- Stage batch slab global -> LDS via the Tensor Data Mover ----
    // One 2-D descriptor: tile = 128x64 f32, row stride 128 elements in memory,
    // HW pad of 4 DWORDs after every 128 DWORDs stored -> LDS stride 132 floats.
    if (tid < 32) {                       // wave 0 issues the DMA (EXEC-independent)
        const uint64_t gaddr = (uint64_t)(uintptr_t)Ab;
        const uint32_t laddr = (uint32_t)(uintptr_t)&lds[0];

        u32x4 g0;
        g0.x = 1u;                                   // count=1 (valid user descriptor)
        g0.y = laddr;                                // lds_addr (bytes)
        g0.z = (uint32_t)(gaddr & 0xFFFFFFFFu);      // global_addr[31:0]
        g0.w = (uint32_t)((gaddr >> 32) & 0x01FFFFFFu) | (2u << 30); // addr[56:32] | type=2

        i32x8 g1;
        g1[0] = (2 << 16)     // data_size = 4 bytes
              | (1 << 20)     // pad_enable
              | (6 << 22)     // pad_interval: pad after 128 DWORDs (= one 512B row)
              | (3 << 25);    // pad_amount: 4 DWORDs (16 B) -> stride 132 floats
        g1[1] = (D_DIM  & 0xFFFF) << 16;  // tensor_dim0 lo16 = 128
        g1[2] = (K_ROWS & 0xFFFF) << 16;  // tensor_dim0 hi16 = 0 | tensor_dim1 lo16 = 64
        g1[3] = (D_DIM  & 0xFFFF) << 16;  // tensor_dim1 hi16 = 0 | tile_dim0 = 128
        g1[4] = K_ROWS;                   // tile_dim1 = 64 | tile_dim2 = 0
        g1[5] = D_DIM;                    // tensor_dim0_stride = 128 elements
        g1[6] = 0;                        // dim0_stride hi | tensor_dim1_stride lo
        g1[7] = 0;                        // tensor_dim1_stride hi (2-D: unused)

        i32x4 gz = {0, 0, 0, 0};          // groups 2/3 unused for a 2-D tensor
#if __clang_major__ >= 23
        i32x8 gz8 = {0, 0, 0, 0, 0, 0, 0, 0};
        __builtin_amdgcn_tensor_load_to_lds(g0, g1, gz, gz, gz8, 0);
#else
        __builtin_amdgcn_tensor_load_to_lds(g0, g1, gz, gz, 0);
#endif
        __builtin_amdgcn_s_wait_tensorcnt(0);   // wave 0 blocks until DMA done
    }
    __syncthreads();                            // releases the other 9 waves
#else
    // ---- Fallback: cooperative coalesced float4 staging ----
    for (int idx = tid; idx < (K_ROWS * D_DIM / 4); idx += NTHREADS) {
        const int row = idx >> 5;          // (idx*4) / 128
        const int col = (idx & 31) << 2;   // (idx*4) % 128
        const float4 v = ((const float4*)Ab)[idx];
        *(float4*)&lds[row * LDS_STRIDE + col] = v;   // 16B-aligned (132*4 % 16 == 0)
    }
    __syncthreads();
#endif

    // ---- Map wave -> lower-triangular tile (ti >= tj) ----
    const int wave = tid >> 5;   // 0..9
    const int lane = tid & 31;
    int ti = 0;
    while ((ti + 1) * (ti + 2) / 2 <= wave) ++ti;       // uniform, <=3 iters
    const int tj = wave - ti * (ti + 1) / 2;

    // Fragment addressing (fp32 WMMA 16x16x4):
    //   A frag: lane m=lane&15 -> row M=m; lanes 0-15 hold K=0,1; lanes 16-31 hold K=2,3
    //   B frag (= A-frag of row-tile tj, since B = Aj^T): identical addressing with N=m
    const int m    = lane & 15;
    const int half = lane >> 4;             // 0: K=0,1   1: K=2,3
    const float* __restrict__ Ai = &lds[(ti * 16 + m) * LDS_STRIDE + 2 * half];
    const float* __restrict__ Aj = &lds[(tj * 16 + m) * LDS_STRIDE + 2 * half];

    v8f c = {0.f, 0.f, 0.f, 0.f, 0.f, 0.f, 0.f, 0.f};

    #pragma unroll
    for (int kk = 0; kk < D_DIM; kk += 4) {
        // Unconditional loads: for diagonal tiles Aj == Ai (same LDS address, broadcast),
        // avoiding any EXEC-mask dance around the WMMA.
        const v2f afrag = *(const v2f*)(Ai + kk);   // ds_load_b64
        const v2f bfrag = *(const v2f*)(Aj + kk);   // ds_load_b64
        // (neg_a, A, neg_b, B, c_mod, C, reuse_a, reuse_b)
        c = __builtin_amdgcn_wmma_f32_16x16x4_f32(false, afrag,
                                                  false, bfrag,
                                                  (short)0, c,
                                                  false, false);
    }

    // ---- Scatter 16x16 tile to packed strict-lower-triangle ----
    // C/D layout: VGPR r, lanes 0-15: (M=r, N=lane); lanes 16-31: (M=r+8, N=lane-16)
    float* __restrict__ outb = out + (size_t)b * NPAIR;
    #pragma unroll
    for (int r = 0; r < 8; ++r) {
        const int gi = ti * 16 + r + 8 * half;
        const int gj = tj * 16 + m;
        if (gi > gj) outb[gi * (gi - 1) / 2 + gj] = c[r];
    }
}

extern "C" void kernel_launch(void* const* d_in, const int* in_sizes, int n_in,
                              void* d_out, int out_size, void* d_ws, size_t ws_size,
                              hipStream_t stream)
{
    const float* A = (const float*)d_in[0];
    float* out     = (float*)d_out;
    const int B    = in_sizes[0] / (K_ROWS * D_DIM);   // 8192 for the reference shapes

    hipLaunchKernelGGL(dot_interaction_wmma_f32,
                       dim3(B), dim3(NTHREADS), 0, stream,
                       A, out);
}